// LocalClassifier_5334349382429
// MI455X (gfx1250) — compile-verified
//
#include <hip/hip_runtime.h>

// CDNA5 / gfx1250: wave32, V_WMMA_F32_16X16X4_F32 (D = A(16x4) * B(4x16) + C)
typedef __attribute__((ext_vector_type(2))) float v2f;
typedef __attribute__((ext_vector_type(8))) float v8f;

constexpr int kB = 32;    // batch
constexpr int kP = 784;   // pixels per image
constexpr int kF = 2048;  // features
constexpr int kSeg = kF / 4;  // 512 features per K-segment wave

// Block = 128 threads = 4 waves = ONE 16-pixel tile, split-K x4:
// wave s reduces features [512*s, 512*(s+1)). Grid (49, 32) -> 6272 waves
// (vs 1568 unsplit): enough concurrency to keep ~14 MB in flight chip-wide
// for the 23.3 TB/s HBM roofline. Partials combine through LDS in a fixed
// order (deterministic), then bias + sigmoid.
//
// K-interleave trick (per 8-feature chunk c): WMMA #0 covers features
// {c,c+1,c+4,c+5}, WMMA #1 covers {c+2,c+3,c+6,c+7}. With the ISA A-layout
// (lane l<16 holds K0,K1; lane l+16 holds K2,K3) every lane's A data for both
// WMMAs is one contiguous float4 at x[p, c + 4*(lane>>4)], ditto for B from
// the LDS-cached weight row. B is nonzero only in column 0 (lanes 0/16),
// enforced with a loop-invariant mask -> v_pk_mul, branchless.
__global__ __launch_bounds__(128) void lc_wmma_kernel(
    const float* __restrict__ x,      // [B*P, F] row-major
    const int* __restrict__ target,   // [B]
    const float* __restrict__ W,      // [C, F]
    const float* __restrict__ bias,   // [C]
    float* __restrict__ out)          // [B*P]
{
    __shared__ __align__(16) float w_lds[kF];       // gathered weight row, 8 KB
    __shared__ float part_lds[4][16];               // per-K-segment partial logits

    const int batch = blockIdx.y;
    const int t     = target[batch];

    // Cooperative coalesced b128 gather of W[target[batch]] into LDS.
    const float* __restrict__ wrow = W + (size_t)t * kF;
    for (int i = threadIdx.x * 4; i < kF; i += 128 * 4) {
        *(float4*)&w_lds[i] = *(const float4*)&wrow[i];
    }
    __syncthreads();

    const int lane = threadIdx.x & 31;
    const int seg  = threadIdx.x >> 5;            // K-segment: 0..3
    const int m    = lane & 15;                   // pixel row within tile
    const int hi   = lane >> 4;                   // half-wave select
    const int pix0 = blockIdx.x << 4;             // base pixel of this tile

    // B column-0 mask: only lanes 0 and 16 carry nonzero B. Branchless.
    const float msk = (m == 0) ? 1.0f : 0.0f;

    // Per-lane A stream over this wave's 512-feature segment.
    const float* __restrict__ xp =
        x + ((size_t)batch * kP + pix0 + m) * kF + seg * kSeg + (hi << 2);
    const float* __restrict__ wp = &w_lds[seg * kSeg + (hi << 2)];

    v8f acc0 = {0.f, 0.f, 0.f, 0.f, 0.f, 0.f, 0.f, 0.f};
    v8f acc1 = {0.f, 0.f, 0.f, 0.f, 0.f, 0.f, 0.f, 0.f};

    // One pipeline stage = 32 features = 4 global b128 + 4 LDS b128 + 8 WMMAs.
    auto step = [&](int c, float4 q0, float4 q1, float4 q2, float4 q3) {
        float4 u0 = *(const float4*)(wp + c);
        float4 u1 = *(const float4*)(wp + c + 8);
        float4 u2 = *(const float4*)(wp + c + 16);
        float4 u3 = *(const float4*)(wp + c + 24);
        v2f a, b;
#define DO_PAIR(q, u)                                                        \
        a.x = q.x;       a.y = q.y;                                          \
        b.x = u.x * msk; b.y = u.y * msk;                                    \
        acc0 = __builtin_amdgcn_wmma_f32_16x16x4_f32(                        \
                   false, a, false, b, (short)0, acc0, false, false);        \
        a.x = q.z;       a.y = q.w;                                          \
        b.x = u.z * msk; b.y = u.w * msk;                                    \
        acc1 = __builtin_amdgcn_wmma_f32_16x16x4_f32(                        \
                   false, a, false, b, (short)0, acc1, false, false);
        DO_PAIR(q0, u0) DO_PAIR(q1, u1) DO_PAIR(q2, u2) DO_PAIR(q3, u3)
#undef DO_PAIR
    };

    // Software-pipelined main loop: prefetch chunk c+32 while computing c,
    // so load latency overlaps the matrix pipe instead of stalling at the
    // loop top (compiler was emitting s_wait_loadcnt right after issue).
    float4 q0 = *(const float4*)(xp + 0);
    float4 q1 = *(const float4*)(xp + 8);
    float4 q2 = *(const float4*)(xp + 16);
    float4 q3 = *(const float4*)(xp + 24);
    for (int c = 0; c < kSeg - 32; c += 32) {
        float4 n0 = *(const float4*)(xp + c + 32);
        float4 n1 = *(const float4*)(xp + c + 40);
        float4 n2 = *(const float4*)(xp + c + 48);
        float4 n3 = *(const float4*)(xp + c + 56);
        step(c, q0, q1, q2, q3);
        q0 = n0; q1 = n1; q2 = n2; q3 = n3;
    }
    step(kSeg - 32, q0, q1, q2, q3);

    // Stash this segment's 16 partial logits (lane 0: pixels 0..7,
    // lane 16: pixels 8..15) into LDS.
    if (m == 0) {
#pragma unroll
        for (int r = 0; r < 8; ++r) {
            part_lds[seg][(hi << 3) + r] = acc0[r] + acc1[r];
        }
    }
    __syncthreads();

    // Fixed-order (deterministic) cross-segment reduction + bias + sigmoid.
    if (threadIdx.x < 16) {
        const int p = threadIdx.x;
        float s = ((part_lds[0][p] + part_lds[1][p]) + part_lds[2][p])
                  + part_lds[3][p];
        float logit = s + bias[t];
        out[(size_t)batch * kP + pix0 + p] = 1.0f / (1.0f + __expf(-logit));
    }
}

extern "C" void kernel_launch(void* const* d_in, const int* in_sizes, int n_in,
                              void* d_out, int out_size, void* d_ws, size_t ws_size,
                              hipStream_t stream) {
    const float* x      = (const float*)d_in[0];  // [B*P, F, 1, 1] f32
    const int*   target = (const int*)d_in[1];    // [B] integer
    const float* W      = (const float*)d_in[2];  // [C, F] f32
    const float* b      = (const float*)d_in[3];  // [C] f32
    float*       out    = (float*)d_out;          // [B*P] f32

    dim3 grid(49, kB);    // 49 pixel tiles * 32 batches = 1568 blocks
    dim3 block(128);      // 4 waves: 1 tile x 4 K-segments
    hipLaunchKernelGGL(lc_wmma_kernel, grid, block, 0, stream,
                       x, target, W, b, out);
}